// SinkhornEMDLoss_13984413516533
// MI455X (gfx1250) — compile-verified
//
#include <hip/hip_runtime.h>

#define B_ 8
#define N_ 1024
#define REG_INV 10.0f      // 1/0.1
#define ITERS 50

typedef __attribute__((ext_vector_type(2))) float v2f;
typedef __attribute__((ext_vector_type(8))) float v8f;

// ---------------------------------------------------------------- init u,v = 0
__global__ void k_init(float* __restrict__ u, float* __restrict__ v) {
  int t = blockIdx.x * blockDim.x + threadIdx.x;
  if (t < B_ * N_) { u[t] = 0.f; v[t] = 0.f; }
}

// --------------------------------------------------- cost tile via WMMA f32 16x16x4
// cost[b][i][j] = relu(|x_i|^2 + |y_j|^2 - 2 x_i . y_j)
// One wave (32 lanes) computes a 16x16 tile; xy via V_WMMA_F32_16X16X4_F32 (K=3 padded to 4).
__global__ __launch_bounds__(32) void k_cost(const float* __restrict__ pred,
                                             const float* __restrict__ gt,
                                             float* __restrict__ cost) {
  __shared__ float sx2[16];
  const int lane = threadIdx.x;     // 0..31
  const int m    = lane & 15;       // row/col index within tile
  const int kh   = lane >> 4;       // K half: 0 -> K{0,1}, 1 -> K{2,3}
  const int i0   = blockIdx.x * 16;
  const int j0   = blockIdx.y * 16;
  const int b    = blockIdx.z;

  const float* P = pred + ((size_t)b * N_ + i0 + m) * 3;
  const float* G = gt   + ((size_t)b * N_ + j0 + m) * 3;
  const float p0 = P[0], p1 = P[1], p2 = P[2];
  const float g0 = G[0], g1 = G[1], g2 = G[2];

  if (lane < 16) sx2[m] = p0 * p0 + p1 * p1 + p2 * p2;   // |x|^2 per tile row
  const float y2 = g0 * g0 + g1 * g1 + g2 * g2;          // |y|^2 for this lane's col

  // A (16x4, MxK): low lanes -> {K0,K1}; high lanes -> {K2, 0(pad)}
  v2f a;  a.x = kh ? p2 : p0;  a.y = kh ? 0.f : p1;
  // B (4x16, KxN): same K-half split, lane = N
  v2f bv; bv.x = kh ? g2 : g0; bv.y = kh ? 0.f : g1;

  v8f acc = {};
  acc = __builtin_amdgcn_wmma_f32_16x16x4_f32(
      /*neg_a=*/false, a, /*neg_b=*/false, bv,
      /*c_mod=*/(short)0, acc, /*reuse_a=*/false, /*reuse_b=*/false);

  __syncthreads();  // sx2 visible

  // D layout: VGPR r -> row (r + 8*kh), col m
  #pragma unroll
  for (int r = 0; r < 8; ++r) {
    const int row = i0 + r + 8 * kh;
    float c = sx2[r + 8 * kh] + y2 - 2.0f * acc[r];
    c = fmaxf(c, 0.0f);
    cost[((size_t)b * N_ + row) * N_ + j0 + m] = c;
  }
}

// --------------------------------------------- row update: u_i = lse_j(-c/REG - v_j)
__global__ __launch_bounds__(256) void k_row(const float* __restrict__ cost,
                                             const float* __restrict__ v,
                                             float* __restrict__ u) {
  __shared__ float red[256];
  const int row = blockIdx.x;                 // 0..B*N-1
  const int b = row >> 10, i = row & (N_ - 1);
  const float* crow = cost + ((size_t)b * N_ + i) * N_;
  const float* vb = v + b * N_;
  const int t = threadIdx.x;

  float s0 = -crow[t      ] * REG_INV - vb[t      ];
  float s1 = -crow[t + 256] * REG_INV - vb[t + 256];
  float s2 = -crow[t + 512] * REG_INV - vb[t + 512];
  float s3 = -crow[t + 768] * REG_INV - vb[t + 768];

  float mx = fmaxf(fmaxf(s0, s1), fmaxf(s2, s3));
  red[t] = mx; __syncthreads();
  for (int st = 128; st > 0; st >>= 1) {
    if (t < st) red[t] = fmaxf(red[t], red[t + st]);
    __syncthreads();
  }
  const float M = red[0]; __syncthreads();

  float sum = __expf(s0 - M) + __expf(s1 - M) + __expf(s2 - M) + __expf(s3 - M);
  red[t] = sum; __syncthreads();
  for (int st = 128; st > 0; st >>= 1) {
    if (t < st) red[t] += red[t + st];
    __syncthreads();
  }
  if (t == 0) u[b * N_ + i] = M + __logf(red[0]);
}

// ---------------- column update, stage 1: online lse partials over 128-row chunks
// grid (8 iChunks, N/256 jChunks, B); thread owns one column j (coalesced reads)
__global__ __launch_bounds__(256) void k_colpart(const float* __restrict__ cost,
                                                 const float* __restrict__ u,
                                                 float* __restrict__ pm,
                                                 float* __restrict__ ps) {
  __shared__ float su[128];
  const int b  = blockIdx.z;
  const int j  = blockIdx.y * 256 + threadIdx.x;
  const int i0 = blockIdx.x * 128;
  if (threadIdx.x < 128) su[threadIdx.x] = u[b * N_ + i0 + threadIdx.x];
  __syncthreads();

  float m = -3.0e38f, s = 0.f;
  const float* cp = cost + ((size_t)b * N_ + i0) * N_ + j;
  #pragma unroll 4
  for (int i = 0; i < 128; ++i) {
    const float val = -cp[0] * REG_INV - su[i];
    cp += N_;
    const float nm = fmaxf(m, val);
    s = s * __expf(m - nm) + __expf(val - nm);
    m = nm;
  }
  const size_t idx = ((size_t)b * 8 + blockIdx.x) * N_ + j;
  pm[idx] = m; ps[idx] = s;
}

// ---------------- column update, stage 2: merge 8 partials -> v_j
__global__ __launch_bounds__(256) void k_colmerge(const float* __restrict__ pm,
                                                  const float* __restrict__ ps,
                                                  float* __restrict__ v) {
  const int t = blockIdx.x * 256 + threadIdx.x;   // 0..B*N-1
  const int b = t >> 10, j = t & (N_ - 1);
  float mk[8], sk[8];
  #pragma unroll
  for (int k = 0; k < 8; ++k) {
    mk[k] = pm[((size_t)b * 8 + k) * N_ + j];
    sk[k] = ps[((size_t)b * 8 + k) * N_ + j];
  }
  float M = mk[0];
  #pragma unroll
  for (int k = 1; k < 8; ++k) M = fmaxf(M, mk[k]);
  float S = 0.f;
  #pragma unroll
  for (int k = 0; k < 8; ++k) S += sk[k] * __expf(mk[k] - M);
  v[b * N_ + j] = M + __logf(S);
}

// ---------------- final loss: per-row partial of sum exp(-c/REG - u_i - v_j) * c
__global__ __launch_bounds__(256) void k_lossrow(const float* __restrict__ cost,
                                                 const float* __restrict__ u,
                                                 const float* __restrict__ v,
                                                 float* __restrict__ part) {
  __shared__ float red[256];
  const int row = blockIdx.x;
  const int b = row >> 10, i = row & (N_ - 1);
  const float* crow = cost + ((size_t)b * N_ + i) * N_;
  const float* vb = v + b * N_;
  const float ui = u[b * N_ + i];
  const int t = threadIdx.x;

  float acc = 0.f;
  #pragma unroll
  for (int k = 0; k < 4; ++k) {
    const int j = t + k * 256;
    const float c = crow[j];
    acc += __expf(-c * REG_INV - ui - vb[j]) * c;
  }
  red[t] = acc; __syncthreads();
  for (int st = 128; st > 0; st >>= 1) {
    if (t < st) red[t] += red[t + st];
    __syncthreads();
  }
  if (t == 0) part[row] = red[0];
}

// ---------------- final reduce (single block, fixed order -> deterministic)
__global__ __launch_bounds__(256) void k_reduce(const float* __restrict__ part,
                                                float* __restrict__ out) {
  __shared__ float red[256];
  float acc = 0.f;
  for (int k = threadIdx.x; k < B_ * N_; k += 256) acc += part[k];
  red[threadIdx.x] = acc; __syncthreads();
  for (int st = 128; st > 0; st >>= 1) {
    if (threadIdx.x < st) red[threadIdx.x] += red[threadIdx.x + st];
    __syncthreads();
  }
  if (threadIdx.x == 0) out[0] = red[0] / ((float)N_ * (float)B_);
}

extern "C" void kernel_launch(void* const* d_in, const int* in_sizes, int n_in,
                              void* d_out, int out_size, void* d_ws, size_t ws_size,
                              hipStream_t stream) {
  const float* pred = (const float*)d_in[0];
  const float* gt   = (const float*)d_in[1];
  float* out = (float*)d_out;

  float* ws   = (float*)d_ws;
  float* cost = ws;                                  // B*N*N      = 8,388,608 f32 (32 MB)
  float* u    = cost + (size_t)B_ * N_ * N_;         // B*N        = 8192
  float* v    = u + (size_t)B_ * N_;                 // B*N        = 8192
  float* pm   = v + (size_t)B_ * N_;                 // B*8*N      = 65536
  float* ps   = pm + (size_t)B_ * 8 * N_;            // B*8*N      = 65536
  float* part = ps + (size_t)B_ * 8 * N_;            // B*N        = 8192

  k_init<<<(B_ * N_ + 255) / 256, 256, 0, stream>>>(u, v);
  k_cost<<<dim3(N_ / 16, N_ / 16, B_), 32, 0, stream>>>(pred, gt, cost);

  for (int it = 0; it < ITERS; ++it) {
    k_row<<<B_ * N_, 256, 0, stream>>>(cost, v, u);
    k_colpart<<<dim3(8, N_ / 256, B_), 256, 0, stream>>>(cost, u, pm, ps);
    k_colmerge<<<(B_ * N_) / 256, 256, 0, stream>>>(pm, ps, v);
  }

  k_lossrow<<<B_ * N_, 256, 0, stream>>>(cost, u, v, part);
  k_reduce<<<1, 256, 0, stream>>>(part, out);
}